// MPNEncoder_69973607186814
// MI455X (gfx1250) — compile-verified
//
#include <hip/hip_runtime.h>

// ---------------------------------------------------------------------------
// D-MPNN encoder for MI455X (gfx1250, wave32).
// All GEMMs are [*,K]@[K,300] fp32 -> V_WMMA_F32_16X16X4_F32 (fp32 matrix pipe).
// W tile is staged in LDS k-pair-interleaved so each B fragment is ONE
// ds_load_b64 straight into the WMMA source register pair.
// ---------------------------------------------------------------------------

typedef __attribute__((ext_vector_type(2))) float v2f;
typedef __attribute__((ext_vector_type(8))) float v8f;

#define HID   300
#define NMOLS 2000
#define BM    32      // rows per block
#define KC    32      // K chunk staged in LDS
#define ALD   36      // LDS A row stride: 36*m mod 64 distinct even values -> no conflicts
#define NPAD  320     // padded N cols in LDS (covers dead tile 19)
#define PWL   672     // pair-row stride (floats): 2*NPAD + 32 ; 672 % 64 == 32
                      // -> the two half-waves of a b64 B-load use disjoint banks

// C[M,300] = A[M,K] @ W[K,300] (+pre)(+bias); optional raw store and/or
// relu store with row-0 zeroing (bond-message padding row).
__global__ __launch_bounds__(256) void gemm300_wmma(
    const float* __restrict__ Amat, int M, int K,
    const float* __restrict__ W,          // K x 300, row-major, compact
    const float* __restrict__ pre,        // nullable: M x 300 addend
    const float* __restrict__ bias,       // nullable: 300
    float* __restrict__ outRaw,           // nullable
    float* __restrict__ outRelu,          // nullable
    int zeroRow0)
{
    __shared__ float lA[BM][ALD];
    __shared__ float lW[KC / 2][PWL];     // lW[k>>1][c*2 + (k&1)] = W[k0+k][c]

    const int tid  = threadIdx.x;
    const int wave = tid >> 5;
    const int lane = tid & 31;
    const int hl   = lane >> 4;          // half-wave select
    const int l16  = lane & 15;
    const int wr   = wave >> 2;          // 0..1 : row tile within block
    const int wc   = wave & 3;           // 0..3 : col tile phase
    const int rowBase = blockIdx.x * BM;

    v8f acc[5];
#pragma unroll
    for (int t = 0; t < 5; ++t) acc[t] = (v8f)0.0f;

    const int nChunks = (K + KC - 1) / KC;
    for (int ch = 0; ch < nChunks; ++ch) {
        const int k0 = ch * KC;
        // ---- stage A tile: BM x KC (zero-pad M/K edges) ----
#pragma unroll
        for (int i = tid; i < BM * KC; i += 256) {
            int r = i >> 5, c = i & 31;           // KC == 32
            int gr = rowBase + r, gk = k0 + c;
            float v = 0.0f;
            if (gr < M && gk < K) v = Amat[(size_t)gr * K + gk];
            lA[r][c] = v;
        }
        // ---- stage W tile, k-pair interleaved: task = (pair p, float4 col grp) ----
        // lW[p][2*c+0] = W[k0+2p][c] ; lW[p][2*c+1] = W[k0+2p+1][c]
#pragma unroll
        for (int i = tid; i < (KC / 2) * 80; i += 256) {   // 16 pairs * 80 f4 groups
            int p  = i / 80, c4 = i % 80;
            int ke = k0 + 2 * p, ko = ke + 1;
            float4 ve = make_float4(0.f, 0.f, 0.f, 0.f);
            float4 vo = make_float4(0.f, 0.f, 0.f, 0.f);
            if (c4 < 75) {
                if (ke < K) ve = *(const float4*)(W + (size_t)ke * HID + c4 * 4);
                if (ko < K) vo = *(const float4*)(W + (size_t)ko * HID + c4 * 4);
            }
            v2f* dst = (v2f*)&lW[p][c4 * 8];   // 4 consecutive {even,odd} b64 pairs
            dst[0] = (v2f){ve.x, vo.x};
            dst[1] = (v2f){ve.y, vo.y};
            dst[2] = (v2f){ve.z, vo.z};
            dst[3] = (v2f){ve.w, vo.w};
        }
        __syncthreads();

        // ---- 8 k-steps of 4, 5 WMMA tiles per wave per step ----
#pragma unroll
        for (int ks = 0; ks < KC / 4; ++ks) {
            const int kk = ks * 4;
            // A frag 16x4 fp32: lanes 0-15 -> K kk,kk+1 ; lanes 16-31 -> kk+2,kk+3
            v2f a = *(const v2f*)&lA[wr * 16 + l16][kk + 2 * hl];
            const int bRow = ks * 2 + hl;     // (kk + 2*hl) >> 1
#pragma unroll
            for (int t = 0; t < 5; ++t) {
                const int c0 = (wc + 4 * t) * 16;
                v2f b = *(const v2f*)&lW[bRow][(c0 + l16) * 2];
                acc[t] = __builtin_amdgcn_wmma_f32_16x16x4_f32(
                    false, a, false, b, (short)0, acc[t], false, false);
            }
        }
        __syncthreads();
    }

    // ---- epilogue: C/D layout VGPR i -> rows (wr*16 + 8*hl + i), col l16 ----
#pragma unroll
    for (int t = 0; t < 5; ++t) {
        const int c0 = (wc + 4 * t) * 16 + l16;
        if (c0 >= HID) continue;
#pragma unroll
        for (int i = 0; i < 8; ++i) {
            const int gr = rowBase + wr * 16 + hl * 8 + i;
            if (gr >= M) continue;
            const size_t idx = (size_t)gr * HID + c0;
            float v = acc[t][i];
            if (pre)  v += pre[idx];
            if (bias) v += bias[c0];
            if (outRaw)  outRaw[idx] = v;
            if (outRelu) outRelu[idx] = (zeroRow0 && gr == 0) ? 0.0f : fmaxf(v, 0.0f);
        }
    }
}

// amsg[a,:] = sum_{j<6} msg[a2b[a,j],:]
__global__ void atom_gather_sum(const float* __restrict__ msg,
                                const int* __restrict__ a2b,
                                float* __restrict__ amsg, int A)
{
    int idx = blockIdx.x * blockDim.x + threadIdx.x;
    if (idx >= A * 75) return;
    int a = idx / 75, h4 = (idx % 75) * 4;
    float4 s = make_float4(0.f, 0.f, 0.f, 0.f);
#pragma unroll
    for (int j = 0; j < 6; ++j) {
        int b = a2b[a * 6 + j];
        float4 v = *(const float4*)(msg + (size_t)b * HID + h4);
        s.x += v.x; s.y += v.y; s.z += v.z; s.w += v.w;
    }
    *(float4*)(amsg + (size_t)a * HID + h4) = s;
}

// m2[b,:] = amsg[b2a[b],:] - msg[b2revb[b],:]
__global__ void bond_gather(const float* __restrict__ amsg,
                            const float* __restrict__ msg,
                            const int* __restrict__ b2a,
                            const int* __restrict__ b2revb,
                            float* __restrict__ m2, int Bn)
{
    int idx = blockIdx.x * blockDim.x + threadIdx.x;
    if (idx >= Bn * 75) return;
    int b = idx / 75, h4 = (idx % 75) * 4;
    int sa = b2a[b], rb = b2revb[b];
    float4 u = *(const float4*)(amsg + (size_t)sa * HID + h4);
    float4 w = *(const float4*)(msg + (size_t)rb * HID + h4);
    float4 r = make_float4(u.x - w.x, u.y - w.y, u.z - w.z, u.w - w.w);
    *(float4*)(m2 + (size_t)b * HID + h4) = r;
}

__global__ void zero_f32(float* __restrict__ p, int n)
{
    int i = blockIdx.x * blockDim.x + threadIdx.x;
    if (i < n) p[i] = 0.0f;
}

__global__ void seg_count(const int* __restrict__ mol_id, float* __restrict__ cnt, int A)
{
    int a = blockIdx.x * blockDim.x + threadIdx.x;
    if (a < A) atomicAdd(&cnt[mol_id[a]], 1.0f);
}

__global__ void seg_sum(const float* __restrict__ hid, const int* __restrict__ mol_id,
                        float* __restrict__ sums, int A)
{
    int idx = blockIdx.x * blockDim.x + threadIdx.x;
    if (idx >= A * 75) return;
    int a = idx / 75, h4 = (idx % 75) * 4;
    int m = mol_id[a];
    float4 v = *(const float4*)(hid + (size_t)a * HID + h4);
    float* dst = sums + (size_t)m * HID + h4;
    atomicAdd(dst + 0, v.x);
    atomicAdd(dst + 1, v.y);
    atomicAdd(dst + 2, v.z);
    atomicAdd(dst + 3, v.w);
}

__global__ void seg_mean(float* __restrict__ sums, const float* __restrict__ cnt, int n)
{
    int i = blockIdx.x * blockDim.x + threadIdx.x;
    if (i >= n) return;
    sums[i] /= fmaxf(cnt[i / HID], 1.0f);
}

// ---------------------------------------------------------------------------

static void run_encoder(const float* f_atoms, const float* f_bonds,
                        const float* W_i, const float* W_h,
                        const float* W_o, const float* b_o,
                        const int* a2b, const int* b2a, const int* b2revb,
                        const int* mol_id,
                        int A, int Bn, int AF, int BF,
                        float* inp, float* msg, float* m2, float* amsg, float* cnt,
                        float* out, hipStream_t stream)
{
    const dim3 blk(256);
    const int gB = (Bn + BM - 1) / BM;
    const int gA = (A + BM - 1) / BM;
    const int gAth = (A * 75 + 255) / 256;
    const int gBth = (Bn * 75 + 255) / 256;

    // inp = f_bonds @ W_i ; msg = relu(inp) with row0 = 0
    gemm300_wmma<<<gB, blk, 0, stream>>>(f_bonds, Bn, BF, W_i,
                                         nullptr, nullptr, inp, msg, 1);
    for (int it = 0; it < 2; ++it) {   // DEPTH-1
        atom_gather_sum<<<gAth, blk, 0, stream>>>(msg, a2b, amsg, A);
        bond_gather<<<gBth, blk, 0, stream>>>(amsg, msg, b2a, b2revb, m2, Bn);
        // msg = relu(inp + m2 @ W_h), row0 = 0
        gemm300_wmma<<<gB, blk, 0, stream>>>(m2, Bn, HID, W_h,
                                             inp, nullptr, nullptr, msg, 1);
    }
    atom_gather_sum<<<gAth, blk, 0, stream>>>(msg, a2b, amsg, A);

    // hid = relu(concat(f_atoms, amsg) @ W_o + b_o), as two accumulating GEMMs
    // pass 1: m2[:A] = f_atoms @ W_o[0:AF]
    gemm300_wmma<<<gA, blk, 0, stream>>>(f_atoms, A, AF, W_o,
                                         nullptr, nullptr, m2, nullptr, 0);
    // pass 2: hid (-> inp buffer) = relu(m2 + amsg @ W_o[AF:] + b_o)
    gemm300_wmma<<<gA, blk, 0, stream>>>(amsg, A, HID, W_o + (size_t)AF * HID,
                                         m2, b_o, nullptr, inp, 0);

    // segment mean readout
    zero_f32<<<(NMOLS * HID + 255) / 256, blk, 0, stream>>>(out, NMOLS * HID);
    zero_f32<<<(NMOLS + 255) / 256, blk, 0, stream>>>(cnt, NMOLS);
    seg_count<<<(A + 255) / 256, blk, 0, stream>>>(mol_id, cnt, A);
    seg_sum<<<gAth, blk, 0, stream>>>(inp, mol_id, out, A);
    seg_mean<<<(NMOLS * HID + 255) / 256, blk, 0, stream>>>(out, cnt, NMOLS * HID);
}

extern "C" void kernel_launch(void* const* d_in, const int* in_sizes, int n_in,
                              void* d_out, int out_size, void* d_ws, size_t ws_size,
                              hipStream_t stream)
{
    const float* mol_f_atoms    = (const float*)d_in[0];
    const float* mol_f_bonds    = (const float*)d_in[1];
    const float* struct_f_atoms = (const float*)d_in[2];
    const float* struct_f_bonds = (const float*)d_in[3];
    const float* W_i1 = (const float*)d_in[4];
    const float* W_i2 = (const float*)d_in[5];
    const float* W_h1 = (const float*)d_in[6];
    const float* W_h2 = (const float*)d_in[7];
    const float* W_o1 = (const float*)d_in[8];
    const float* b_o1 = (const float*)d_in[9];
    const float* W_o2 = (const float*)d_in[10];
    const float* b_o2 = (const float*)d_in[11];
    const int* mol_a2b       = (const int*)d_in[12];
    const int* mol_b2a       = (const int*)d_in[13];
    const int* mol_b2revb    = (const int*)d_in[14];
    const int* mol_mol_id    = (const int*)d_in[15];
    const int* struct_a2b    = (const int*)d_in[16];
    const int* struct_b2a    = (const int*)d_in[17];
    const int* struct_b2revb = (const int*)d_in[18];
    const int* struct_mol_id = (const int*)d_in[19];

    const int A  = in_sizes[15];            // 100001 atoms
    const int Bn = in_sizes[13];            // 200001 bonds
    const int BF = in_sizes[4] / HID;       // 147
    const int AF = in_sizes[8] / HID - HID; // 133

    // workspace layout (fp32, all 16B aligned since 300 % 4 == 0)
    float* ws   = (float*)d_ws;
    float* inp  = ws;                                 // Bn*300
    float* msg  = inp + (size_t)Bn * HID;             // Bn*300
    float* m2   = msg + (size_t)Bn * HID;             // Bn*300
    float* amsg = m2 + (size_t)Bn * HID;              // A*300
    float* cnt  = amsg + (size_t)A * HID;             // NMOLS

    float* out = (float*)d_out;                       // [2000*300 | 2000*300]

    run_encoder(mol_f_atoms, mol_f_bonds, W_i1, W_h1, W_o1, b_o1,
                mol_a2b, mol_b2a, mol_b2revb, mol_mol_id,
                A, Bn, AF, BF, inp, msg, m2, amsg, cnt,
                out, stream);
    run_encoder(struct_f_atoms, struct_f_bonds, W_i2, W_h2, W_o2, b_o2,
                struct_a2b, struct_b2a, struct_b2revb, struct_mol_id,
                A, Bn, AF, BF, inp, msg, m2, amsg, cnt,
                out + NMOLS * HID, stream);
}